// YOLOLoss_16346645529064
// MI455X (gfx1250) — compile-verified
//
#include <hip/hip_runtime.h>
#include <cstdint>

// ---------------------------------------------------------------------------
// YOLO loss on MI455X (gfx1250): pure streaming reduction.
//   88.2 MB read once @ 23.3 TB/s => ~3.8 us floor; FLOPs negligible.
//   CDNA5 path: straight-line global_load_async_to_lds_b128 with literal
//   offset immediates (ASYNCcnt), conflict-free LDS gathers, ds_swizzle
//   wave32 butterfly, per-block atomics + last-block finalize (2 graph nodes).
// ---------------------------------------------------------------------------

#define NC          20
#define PRED_STRIDE 30          // BB*5 + C
#define TARG_STRIDE 25          // C + 5
#define TILE        256         // cells per workgroup
#define THREADS     256         // 8 waves (wave32)
#define L_COORD     5.0f
#define L_NOOBJ     0.5f

// pred tile = 256*30*4 = 30720 B = 7.5 * (256 lanes * 16B)
// targ tile = 256*25*4 = 25600 B = 6.25 * (256 lanes * 16B)

// Async global->LDS, 16B/lane. INST_OFFSET is added to BOTH the global
// address and the LDS address, so one literal offset advances both tiles.
#define ALD16(ldsoff, gptr, OFF)                                            \
    asm volatile("global_load_async_to_lds_b128 %0, %1, off offset:" #OFF   \
                 :: "v"(ldsoff), "v"(gptr) : "memory")

// wave32 butterfly add via DS_SWIZZLE_B32 (group-of-32: and=0x1f, xor=m)
#define SWZ_XOR(v, m) \
    __int_as_float(__builtin_amdgcn_ds_swizzle(__float_as_int(v), 0x1f | ((m) << 10)))

__device__ __forceinline__ float wave_sum(float v) {
    v += SWZ_XOR(v, 1);
    v += SWZ_XOR(v, 2);
    v += SWZ_XOR(v, 4);
    v += SWZ_XOR(v, 8);
    v += SWZ_XOR(v, 16);
    return v;
}

__device__ __forceinline__ float iou_f(float px, float py, float pw, float ph,
                                       float tx, float ty, float tw, float th) {
    float ax1 = px - 0.5f * pw, ax2 = px + 0.5f * pw;
    float ay1 = py - 0.5f * ph, ay2 = py + 0.5f * ph;
    float bx1 = tx - 0.5f * tw, bx2 = tx + 0.5f * tw;
    float by1 = ty - 0.5f * th, by2 = ty + 0.5f * th;
    float iw = fmaxf(fminf(ax2, bx2) - fmaxf(ax1, bx1), 0.0f);
    float ih = fmaxf(fminf(ay2, by2) - fmaxf(ay1, by1), 0.0f);
    float inter = iw * ih;
    float uni = fabsf(pw * ph) + fabsf(tw * th) - inter;
    return inter / (uni + 1e-6f);
}

__global__ __launch_bounds__(THREADS)
void yolo_loss_kernel(const float* __restrict__ pred,
                      const float* __restrict__ targ,
                      float* __restrict__ acc,
                      float* __restrict__ out) {
    __shared__ __align__(16) float sp[TILE * PRED_STRIDE];   // 30720 B
    __shared__ __align__(16) float st[TILE * TARG_STRIDE];   // 25600 B
    __shared__ float red[3 * (THREADS / 32)];

    const int tid = threadIdx.x;
    const size_t base = (size_t)blockIdx.x * TILE;

    // Per-thread staging bases: global and LDS advance together via offset:
    const char* gpt = (const char*)(pred + base * PRED_STRIDE) + (size_t)tid * 16;
    const char* gtt = (const char*)(targ + base * TARG_STRIDE) + (size_t)tid * 16;
    const uint32_t spt = (uint32_t)(uintptr_t)sp + (uint32_t)tid * 16u;
    const uint32_t stt = (uint32_t)(uintptr_t)st + (uint32_t)tid * 16u;

    // pred: 7 full sweeps + half sweep (waves 0-3 only: uniform branch)
    ALD16(spt, gpt, 0);
    ALD16(spt, gpt, 4096);
    ALD16(spt, gpt, 8192);
    ALD16(spt, gpt, 12288);
    ALD16(spt, gpt, 16384);
    ALD16(spt, gpt, 20480);
    ALD16(spt, gpt, 24576);
    if (tid < 128) { ALD16(spt, gpt, 28672); }

    // targ: 6 full sweeps + quarter sweep (waves 0-1 only: uniform branch)
    ALD16(stt, gtt, 0);
    ALD16(stt, gtt, 4096);
    ALD16(stt, gtt, 8192);
    ALD16(stt, gtt, 12288);
    ALD16(stt, gtt, 16384);
    ALD16(stt, gtt, 20480);
    if (tid < 64) { ALD16(stt, gtt, 24576); }

    asm volatile("s_wait_asynccnt 0" ::: "memory");
    __syncthreads();

    // One cell per thread; strides 30/25 are bank-conflict-free mod 64 banks.
    const float* cp = sp + tid * PRED_STRIDE;
    const float* ct = st + tid * TARG_STRIDE;

    float cls_sq = 0.0f;
#pragma unroll
    for (int c = 0; c < NC; ++c) {
        float d = cp[c] - ct[c];
        cls_sq = fmaf(d, d, cls_sq);
    }

    const float tconf = ct[20];
    const float tx = ct[21], ty = ct[22], tw = ct[23], th = ct[24];

    const float c0 = cp[20], x0 = cp[21], y0 = cp[22], w0 = cp[23], h0 = cp[24];
    const float c1 = cp[25], x1 = cp[26], y1 = cp[27], w1 = cp[28], h1 = cp[29];

    const float iou0 = iou_f(x0, y0, w0, h0, tx, ty, tw, th);
    const float iou1 = iou_f(x1, y1, w1, h1, tx, ty, tw, th);

    const bool take1 = iou1 > iou0;               // argmax, first-wins on tie
    const float iou_best = take1 ? iou1 : iou0;
    const float pc = take1 ? c1 : c0;
    const float px = take1 ? x1 : x0;
    const float py = take1 ? y1 : y0;
    const float pw = take1 ? w1 : w0;
    const float ph = take1 ? h1 : h0;

    const float objf = (tconf == 1.0f) ? 1.0f : 0.0f;

    const float dx = tx - px, dy = ty - py;
    const float center = dx * dx + dy * dy;
    const float sw = sqrtf(tw) - sqrtf(fabsf(pw));
    const float sh = sqrtf(th) - sqrtf(fabsf(ph));
    const float wh = sw * sw + sh * sh;
    const float dc = tconf - pc;
    const float conf_sq = dc * dc;

    float lossv = objf * (L_COORD * (center + wh) + conf_sq + cls_sq)
                + (1.0f - objf) * L_NOOBJ * conf_sq;
    float iouv = iou_best * objf;
    float cntv = objf;

    // wave32 butterfly then cross-wave LDS reduce
    lossv = wave_sum(lossv);
    iouv  = wave_sum(iouv);
    cntv  = wave_sum(cntv);

    const int lane = tid & 31;
    const int wave = tid >> 5;
    if (lane == 0) {
        red[wave]      = lossv;
        red[8 + wave]  = iouv;
        red[16 + wave] = cntv;
    }
    __syncthreads();

    if (tid == 0) {
        float a = 0.0f, b = 0.0f, c = 0.0f;
#pragma unroll
        for (int w = 0; w < THREADS / 32; ++w) {
            a += red[w];
            b += red[8 + w];
            c += red[16 + w];
        }
        atomicAdd(acc + 0, a);   // loss sum
        atomicAdd(acc + 1, b);   // iou*obj sum
        atomicAdd(acc + 2, c);   // obj count
        __threadfence();
        unsigned prev = atomicAdd((unsigned*)(acc + 3), 1u);
        if (prev == gridDim.x - 1u) {
            // last block: coherent re-read of the accumulators, write outputs
            float ls = atomicAdd(acc + 0, 0.0f);
            float is = atomicAdd(acc + 1, 0.0f);
            float cs = atomicAdd(acc + 2, 0.0f);
            out[0] = ls;
            out[1] = is / fmaxf(cs, 1.0f);
        }
    }
}

__global__ void yolo_init_acc(float* __restrict__ acc) {
    // zero loss/iou/cnt accumulators + ticket counter (ws is poisoned 0xAA
    // and never re-poisoned between graph replays -> must re-init every call)
    if (threadIdx.x < 4) acc[threadIdx.x] = 0.0f;
}

extern "C" void kernel_launch(void* const* d_in, const int* in_sizes, int n_in,
                              void* d_out, int out_size, void* d_ws, size_t ws_size,
                              hipStream_t stream) {
    (void)n_in; (void)out_size; (void)ws_size;
    const float* pred = (const float*)d_in[0];
    const float* targ = (const float*)d_in[1];
    float* acc = (float*)d_ws;          // [loss, iou_sum, cnt, ticket]
    float* out = (float*)d_out;         // [loss, avg_iou]

    const int cells  = in_sizes[0] / PRED_STRIDE;  // 8192*7*7 = 401408
    const int blocks = cells / TILE;               // 1568 (exact)

    yolo_init_acc<<<1, 32, 0, stream>>>(acc);
    yolo_loss_kernel<<<blocks, THREADS, 0, stream>>>(pred, targ, acc, out);
}